// DMDNet_26680336843458
// MI455X (gfx1250) — compile-verified
//
#include <hip/hip_runtime.h>
#include <hip/hip_bf16.h>

// Problem constants (fixed by the reference harness).
#define MDIM 1024      // latent dim m (= n contraction per window slot)
#define BDIM 256       // batch
#define LWIN 8         // window length
#define PRED 8         // predict length
#define NSLOT (LWIN + PRED)

typedef __attribute__((ext_vector_type(2))) float v2f;
typedef __attribute__((ext_vector_type(8))) float v8f;

__device__ __forceinline__ v8f wmma_f32_k4(v2f a, v2f b, v8f c) {
    // V_WMMA_F32_16X16X4_F32 : D = A(16x4) x B(4x16) + C(16x16), exact fp32.
    return __builtin_amdgcn_wmma_f32_16x16x4_f32(
        /*neg_a=*/false, a, /*neg_b=*/false, b,
        /*c_mod=*/(short)0, c, /*reuse_a=*/false, /*reuse_b=*/false);
}

// ---------------------------------------------------------------------------
// Build AgT: AgT[p][n][m] = A[idx(p)][m][n], idx(0)=0, idx(p)=L-p.
// Stored n-pair interleaved: elem(p,n,m) at p*M*M + (n>>1)*(2*M) + 2*m + (n&1)
// so a lane's (K, K+1) A-fragment pair is one contiguous float2.
// ---------------------------------------------------------------------------
__global__ void dmd_build_agt(const float* __restrict__ Ar,
                              const float* __restrict__ Ai,
                              float* __restrict__ agtr,
                              float* __restrict__ agti) {
    size_t i = (size_t)blockIdx.x * blockDim.x + threadIdx.x; // over 8*1024*1024
    if (i >= (size_t)LWIN * MDIM * MDIM) return;
    int n = (int)(i & (MDIM - 1));
    int m = (int)((i >> 10) & (MDIM - 1));
    int p = (int)(i >> 20);
    int l = (p == 0) ? 0 : (LWIN - p);
    size_t src = (size_t)l * MDIM * MDIM + (size_t)m * MDIM + n;
    size_t dst = (size_t)p * MDIM * MDIM + (size_t)(n >> 1) * (2 * MDIM)
               + 2 * m + (n & 1);
    agtr[dst] = Ar[src];
    agti[dst] = Ai[src];
}

// ---------------------------------------------------------------------------
// Init state slots 0..7 from x[b][s][n]; imag = 0.
// State plane layout: elem(s,n,b) at s*M*B + (n>>1)*(2*B) + 2*b + (n&1).
// ---------------------------------------------------------------------------
__global__ void dmd_init_state(const float* __restrict__ x,
                               float* __restrict__ stR,
                               float* __restrict__ stI) {
    size_t i = (size_t)blockIdx.x * blockDim.x + threadIdx.x; // over 8*1024*256
    if (i >= (size_t)LWIN * MDIM * BDIM) return;
    int b = (int)(i & (BDIM - 1));
    int n = (int)((i >> 8) & (MDIM - 1));
    int s = (int)(i >> 18);
    float v = x[(size_t)b * (LWIN * MDIM) + (size_t)s * MDIM + n];
    size_t dst = (size_t)s * MDIM * BDIM + (size_t)(n >> 1) * (2 * BDIM)
               + 2 * b + (n & 1);
    stR[dst] = v;
    stI[dst] = 0.0f;
}

// ---------------------------------------------------------------------------
// One recurrence step, split-K over the 8 window positions (blockIdx.z = p).
// One wave per block (EXEC all ones, as WMMA requires).
// Wave computes a 64(m) x 16(b) complex tile partial for its p:
//   rr += AgR*UR, ii += AgI*UI, ri += AgR*UI, ir += AgI*UR
// then writes partR = rr - ii, partI = ri + ir into partials[p][m][b].
// ---------------------------------------------------------------------------
__global__ __launch_bounds__(32) void dmd_gemm_step(
    const float* __restrict__ agtr, const float* __restrict__ agti,
    const float* __restrict__ stR,  const float* __restrict__ stI,
    float* __restrict__ partR, float* __restrict__ partI, int t) {
    const int lane = threadIdx.x & 31;
    const int g  = lane >> 4;   // lane half selects K pair (K+2)
    const int lm = lane & 15;   // row (A) / column (B,D) within tile
    const int p  = blockIdx.z;
    const int m0 = blockIdx.y * 64;
    const int b0 = blockIdx.x * 16;
    const int s  = t + p;       // window slot for this p

    const float* aR = agtr + (size_t)p * MDIM * MDIM;
    const float* aI = agti + (size_t)p * MDIM * MDIM;
    const float* uR = stR + (size_t)s * MDIM * BDIM;
    const float* uI = stI + (size_t)s * MDIM * BDIM;

    v8f c_rr[4] = {}, c_ii[4] = {}, c_ri[4] = {}, c_ir[4] = {};

    for (int k = 0; k < MDIM; k += 4) {
        const int np = (k + 2 * g) >> 1;  // interleaved n-pair index
        // B fragments (state), one coalesced b64 load per plane.
        const float2 bR2 = *(const float2*)(uR + (size_t)np * (2 * BDIM) + 2 * (b0 + lm));
        const float2 bI2 = *(const float2*)(uI + (size_t)np * (2 * BDIM) + 2 * (b0 + lm));
        v2f fbR = { bR2.x, bR2.y };
        v2f fbI = { bI2.x, bI2.y };
#pragma unroll
        for (int mt = 0; mt < 4; ++mt) {
            const int m = m0 + mt * 16 + lm;
            const float2 aR2 = *(const float2*)(aR + (size_t)np * (2 * MDIM) + 2 * m);
            const float2 aI2 = *(const float2*)(aI + (size_t)np * (2 * MDIM) + 2 * m);
            v2f faR = { aR2.x, aR2.y };
            v2f faI = { aI2.x, aI2.y };
            c_rr[mt] = wmma_f32_k4(faR, fbR, c_rr[mt]);
            c_ii[mt] = wmma_f32_k4(faI, fbI, c_ii[mt]);
            c_ri[mt] = wmma_f32_k4(faR, fbI, c_ri[mt]);
            c_ir[mt] = wmma_f32_k4(faI, fbR, c_ir[mt]);
        }
    }

    // D layout (16x16 f32): vgpr r -> row m = r + 8*g ; lane lm -> col b.
#pragma unroll
    for (int mt = 0; mt < 4; ++mt) {
#pragma unroll
        for (int r = 0; r < 8; ++r) {
            const int m = m0 + mt * 16 + r + 8 * g;
            const size_t o = (size_t)p * MDIM * BDIM + (size_t)m * BDIM + b0 + lm;
            partR[o] = c_rr[mt][r] - c_ii[mt][r];
            partI[o] = c_ri[mt][r] + c_ir[mt][r];
        }
    }
}

// ---------------------------------------------------------------------------
// Deterministic split-K reduction: sum 8 partials -> new state slot (8+t).
// ---------------------------------------------------------------------------
__global__ void dmd_reduce_step(const float* __restrict__ partR,
                                const float* __restrict__ partI,
                                float* __restrict__ stR,
                                float* __restrict__ stI, int t) {
    int i = blockIdx.x * blockDim.x + threadIdx.x; // over M*B
    if (i >= MDIM * BDIM) return;
    int m = i >> 8;
    int b = i & (BDIM - 1);
    float sr = 0.0f, si = 0.0f;
#pragma unroll
    for (int p = 0; p < LWIN; ++p) {
        sr += partR[(size_t)p * MDIM * BDIM + i];
        si += partI[(size_t)p * MDIM * BDIM + i];
    }
    size_t dst = (size_t)(LWIN + t) * MDIM * BDIM
               + (size_t)(m >> 1) * (2 * BDIM) + 2 * b + (m & 1);
    stR[dst] = sr;
    stI[dst] = si;
}

// ---------------------------------------------------------------------------
// out[b][l][m] = real(state slot 8+l at (m, b)).
// ---------------------------------------------------------------------------
__global__ void dmd_write_out(const float* __restrict__ stR,
                              float* __restrict__ out) {
    size_t i = (size_t)blockIdx.x * blockDim.x + threadIdx.x; // over B*L*M
    if (i >= (size_t)BDIM * LWIN * MDIM) return;
    int m = (int)(i & (MDIM - 1));
    int l = (int)((i >> 10) & (LWIN - 1));
    int b = (int)(i >> 13);
    size_t src = (size_t)(LWIN + l) * MDIM * BDIM
               + (size_t)(m >> 1) * (2 * BDIM) + 2 * b + (m & 1);
    out[i] = stR[src];
}

extern "C" void kernel_launch(void* const* d_in, const int* in_sizes, int n_in,
                              void* d_out, int out_size, void* d_ws, size_t ws_size,
                              hipStream_t stream) {
    const float* x  = (const float*)d_in[0];  // [256, 8, 1024] f32
    const float* Ar = (const float*)d_in[1];  // [8, 1024, 1024] f32
    const float* Ai = (const float*)d_in[2];  // [8, 1024, 1024] f32
    // d_in[3] = predict_length (device scalar); fixed at 8 by the harness.

    float* ws = (float*)d_ws;
    const size_t AGT  = (size_t)LWIN * MDIM * MDIM;   // 8M floats each
    const size_t STPL = (size_t)NSLOT * MDIM * BDIM;  // 4M floats each
    const size_t PART = (size_t)LWIN * MDIM * BDIM;   // 2M floats each
    float* agtr = ws;
    float* agti = agtr + AGT;
    float* stR  = agti + AGT;
    float* stI  = stR + STPL;
    float* pR   = stI + STPL;
    float* pI   = pR + PART;
    // total ws use: 28M floats = 112 MB

    dmd_build_agt<<<(unsigned)(AGT / 256), 256, 0, stream>>>(Ar, Ai, agtr, agti);
    dmd_init_state<<<(unsigned)(((size_t)LWIN * MDIM * BDIM) / 256), 256, 0, stream>>>(x, stR, stI);

    for (int t = 0; t < PRED; ++t) {
        dmd_gemm_step<<<dim3(BDIM / 16, MDIM / 64, LWIN), 32, 0, stream>>>(
            agtr, agti, stR, stI, pR, pI, t);
        dmd_reduce_step<<<(MDIM * BDIM) / 256, 256, 0, stream>>>(pR, pI, stR, stI, t);
    }

    dmd_write_out<<<(unsigned)(((size_t)BDIM * LWIN * MDIM) / 256), 256, 0, stream>>>(
        stR, (float*)d_out);
}